// GCN_10943576670340
// MI455X (gfx1250) — compile-verified
//
#include <hip/hip_runtime.h>

#define N_NODES   100000
#define N_EDGES   1600000
#define NF        128
#define N_GRAPHS  256
#define N_CLASSES 2
#define BN_EPS    1e-5f
#define LDSTRIDE  132   // padded row stride (dwords): 16B-aligned rows, conflict-free b64 reads
#define MT        32    // output rows per block (two 16-row WMMA tiles per wave)

typedef float v2f __attribute__((ext_vector_type(2)));
typedef float v8f __attribute__((ext_vector_type(8)));

// ---------------------------------------------------------------- fill ----
__global__ void fill_kernel(float* __restrict__ p, int n, float v) {
    int i = blockIdx.x * blockDim.x + threadIdx.x;
    if (i < n) p[i] = v;
}

// ------------------------------------------------------- weight repack ----
// P[p][n] = (W[2p][n], W[2p+1][n]) as float2 -> B fragment is one b64 load
__global__ void pack_kernel(const float* __restrict__ W, float* __restrict__ P) {
    int i = blockIdx.x * blockDim.x + threadIdx.x;   // [0, 64*128)
    if (i >= (NF / 2) * NF) return;
    int p = i >> 7;
    int n = i & 127;
    float2 v = make_float2(W[(2 * p) * NF + n], W[(2 * p + 1) * NF + n]);
    *(float2*)(P + i * 2) = v;
}

// ---------------------------------------------- edge scatter (aggregate) ----
// agg[dst] += feat[src]; one thread per (edge, 4-feature chunk)
__global__ void scatter_kernel(const float* __restrict__ feat,
                               const int* __restrict__ src,
                               const int* __restrict__ dst,
                               float* __restrict__ agg) {
    int i = blockIdx.x * blockDim.x + threadIdx.x;     // [0, N_EDGES*32)
    int e = i >> 5;
    int c = (i & 31) * 4;
    if (e >= N_EDGES) return;
    int s = src[e];
    int d = dst[e];
    const float4 v = *(const float4*)(feat + s * NF + c);
    float* o = agg + d * NF + c;
    atomicAdd(o + 0, v.x);
    atomicAdd(o + 1, v.y);
    atomicAdd(o + 2, v.z);
    atomicAdd(o + 3, v.w);
}

// ------------------------------------------------------- WMMA dual GEMM ----
// H = A0 @ B0 + A1 @ B1 + bias  (optional ReLU)
// Block = 256 threads = 8 waves; block -> 32 rows, wave w -> cols [16w,16w+16)
// Each wave holds two 16x16 accumulators (row tiles), sharing B fragments.
// A tiles async-staged global->LDS (B128, coalesced), frags via ds_load_b64.
// B matrices are pre-packed so a fragment is one contiguous b64 load.
__global__ __launch_bounds__(256) void gemm_kernel(
    const float* __restrict__ A0, const float* __restrict__ A1,
    const float* __restrict__ B0p, const float* __restrict__ B1p,
    const float* __restrict__ bias, float* __restrict__ H, int relu) {
    __shared__ float ldsA[2 * MT * LDSTRIDE];   // ~33 KB: two 32x128 tiles, padded
    const int lane = threadIdx.x & 31;
    const int wave = threadIdx.x >> 5;
    const int row0 = blockIdx.x * MT;           // 3125 blocks * 32 = 100000 exactly
    const int n0   = wave * 16;
    const int m    = lane & 15;
    const int half = lane >> 4;

    // ---- async-stage both 32x128 A tiles into LDS ----
    {
        const float* g0 = A0 + row0 * NF;       // contiguous 16 KB region
        const float* g1 = A1 + row0 * NF;
#pragma unroll
        for (int it = 0; it < 4; ++it) {
            int chunk = threadIdx.x + it * 256;  // [0,1024): 16-byte chunks
            int r  = chunk >> 5;                 // tile row
            int cc = chunk & 31;                 // 16B chunk within row
            unsigned l0 = (unsigned)(uintptr_t)&ldsA[r * LDSTRIDE + cc * 4];
            unsigned l1 = (unsigned)(uintptr_t)&ldsA[(MT + r) * LDSTRIDE + cc * 4];
            unsigned long long ga0 = (unsigned long long)(uintptr_t)(g0 + chunk * 4);
            unsigned long long ga1 = (unsigned long long)(uintptr_t)(g1 + chunk * 4);
            asm volatile("global_load_async_to_lds_b128 %0, %1, off"
                         :: "v"(l0), "v"(ga0) : "memory");
            asm volatile("global_load_async_to_lds_b128 %0, %1, off"
                         :: "v"(l1), "v"(ga1) : "memory");
        }
        asm volatile("s_wait_asynccnt 0x0" ::: "memory");
        __syncthreads();
    }

    v8f acc0 = {};   // rows [row0,    row0+16)
    v8f acc1 = {};   // rows [row0+16, row0+32)
    const float* la0t0 = &ldsA[m * LDSTRIDE + 2 * half];
    const float* la0t1 = &ldsA[(16 + m) * LDSTRIDE + 2 * half];
    const float* la1t0 = &ldsA[(MT + m) * LDSTRIDE + 2 * half];
    const float* la1t1 = &ldsA[(MT + 16 + m) * LDSTRIDE + 2 * half];
    // packed weights: fragment(k) = P[(k/2 + half)*128 + n0 + m] (float2)
    const float* bp0 = B0p + (half * NF + n0 + m) * 2;
    const float* bp1 = B1p + (half * NF + n0 + m) * 2;

#pragma unroll
    for (int k = 0; k < NF; k += 4) {
        v2f b0  = *(const v2f*)(bp0 + k * NF);   // coalesced global b64
        v2f b1  = *(const v2f*)(bp1 + k * NF);
        v2f a00 = *(const v2f*)(la0t0 + k);      // ds_load_b64 x4
        v2f a01 = *(const v2f*)(la0t1 + k);
        v2f a10 = *(const v2f*)(la1t0 + k);
        v2f a11 = *(const v2f*)(la1t1 + k);
        acc0 = __builtin_amdgcn_wmma_f32_16x16x4_f32(false, a00, false, b0,
                                                     (short)0, acc0, false, false);
        acc1 = __builtin_amdgcn_wmma_f32_16x16x4_f32(false, a01, false, b0,
                                                     (short)0, acc1, false, false);
        acc0 = __builtin_amdgcn_wmma_f32_16x16x4_f32(false, a10, false, b1,
                                                     (short)0, acc0, false, false);
        acc1 = __builtin_amdgcn_wmma_f32_16x16x4_f32(false, a11, false, b1,
                                                     (short)0, acc1, false, false);
    }

    const float bc = bias[n0 + m];
#pragma unroll
    for (int v = 0; v < 8; ++v) {
        float v0 = acc0[v] + bc;
        float v1 = acc1[v] + bc;
        if (relu) {
            v0 = fmaxf(v0, 0.0f);
            v1 = fmaxf(v1, 0.0f);
        }
        H[(row0 + v + 8 * half) * NF + n0 + m]      = v0;
        H[(row0 + 16 + v + 8 * half) * NF + n0 + m] = v1;
    }
}

// --------------------------------------------- per-column sum / sum-of-sq ----
__global__ __launch_bounds__(256) void stats_kernel(const float* __restrict__ H,
                                                    float* __restrict__ colsum,
                                                    float* __restrict__ colsumsq) {
    __shared__ float ssum[256];
    __shared__ float ssq[256];
    const int col    = threadIdx.x & 127;
    const int halfid = threadIdx.x >> 7;   // 0 or 1
    const int rowsPer = (N_NODES + gridDim.x - 1) / gridDim.x;
    int r0 = blockIdx.x * rowsPer;
    int r1 = r0 + rowsPer;
    if (r1 > N_NODES) r1 = N_NODES;
    float s = 0.0f, q = 0.0f;
    for (int r = r0 + halfid; r < r1; r += 2) {
        float v = H[r * NF + col];
        s += v;
        q += v * v;
    }
    ssum[threadIdx.x] = s;
    ssq[threadIdx.x]  = q;
    __syncthreads();
    if (halfid == 0) {
        atomicAdd(&colsum[col],   ssum[col] + ssum[col + 128]);
        atomicAdd(&colsumsq[col], ssq[col] + ssq[col + 128]);
    }
}

// ----------------------------------------------------------- BN normalize ----
__global__ void bn_kernel(const float* __restrict__ H,
                          const float* __restrict__ colsum,
                          const float* __restrict__ colsumsq,
                          const float* __restrict__ gamma,
                          const float* __restrict__ beta,
                          float* __restrict__ O) {
    int i = blockIdx.x * blockDim.x + threadIdx.x;
    if (i >= N_NODES * NF) return;
    int col = i & 127;
    const float invN = 1.0f / (float)N_NODES;
    float mean = colsum[col] * invN;
    float var  = colsumsq[col] * invN - mean * mean;
    float inv  = rsqrtf(var + BN_EPS);
    O[i] = (H[i] - mean) * inv * gamma[col] + beta[col];
}

// ------------------------------------------------------- global mean pool ----
__global__ void pool_kernel(const float* __restrict__ hn,
                            const int* __restrict__ batch,
                            float* __restrict__ gsum,
                            float* __restrict__ gcnt) {
    int i = blockIdx.x * blockDim.x + threadIdx.x;   // [0, N_NODES*32)
    int node = i >> 5;
    int c    = (i & 31) * 4;
    if (node >= N_NODES) return;
    int g = batch[node];
    const float4 v = *(const float4*)(hn + node * NF + c);
    float* o = gsum + g * NF + c;
    atomicAdd(o + 0, v.x);
    atomicAdd(o + 1, v.y);
    atomicAdd(o + 2, v.z);
    atomicAdd(o + 3, v.w);
    if (c == 0) atomicAdd(&gcnt[g], 1.0f);
}

// --------------------------------------------------------- tiny classifier ----
__global__ void final_kernel(const float* __restrict__ gsum,
                             const float* __restrict__ gcnt,
                             const float* __restrict__ Wlin,
                             const float* __restrict__ blin,
                             float* __restrict__ out) {
    int g = blockIdx.x * blockDim.x + threadIdx.x;
    if (g >= N_GRAPHS) return;
    float inv = 1.0f / fmaxf(gcnt[g], 1.0f);
    float acc0 = blin[0];
    float acc1 = blin[1];
#pragma unroll 4
    for (int c = 0; c < NF; ++c) {
        float p = gsum[g * NF + c] * inv;
        acc0 += p * Wlin[c * N_CLASSES + 0];
        acc1 += p * Wlin[c * N_CLASSES + 1];
    }
    out[g * N_CLASSES + 0] = acc0;
    out[g * N_CLASSES + 1] = acc1;
}

// =========================================================== orchestration ==
extern "C" void kernel_launch(void* const* d_in, const int* in_sizes, int n_in,
                              void* d_out, int out_size, void* d_ws, size_t ws_size,
                              hipStream_t stream) {
    const float* x     = (const float*)d_in[0];
    const int*   edge  = (const int*)d_in[1];
    const int*   batch = (const int*)d_in[2];
    const int*   src   = edge;
    const int*   dst   = edge + N_EDGES;

    const float* Wr[3]    = {(const float*)d_in[3], (const float*)d_in[8],  (const float*)d_in[13]};
    const float* bb[3]    = {(const float*)d_in[4], (const float*)d_in[9],  (const float*)d_in[14]};
    const float* Wroot[3] = {(const float*)d_in[5], (const float*)d_in[10], (const float*)d_in[15]};
    const float* gm[3]    = {(const float*)d_in[6], (const float*)d_in[11], (const float*)d_in[16]};
    const float* be[3]    = {(const float*)d_in[7], (const float*)d_in[12], (const float*)d_in[17]};
    const float* Wlin = (const float*)d_in[18];
    const float* blin = (const float*)d_in[19];
    float* out = (float*)d_out;

    // workspace layout (floats)
    float* ws = (float*)d_ws;
    const int NE = N_NODES * NF;            // 12,800,000
    float* agg      = ws;                   // [N_NODES, 128]
    float* h        = agg + NE;             // [N_NODES, 128]
    float* hn       = h + NE;               // [N_NODES, 128]
    float* colsum   = hn + NE;              // [128]
    float* colsumsq = colsum + NF;          // [128]
    float* gsum     = colsumsq + NF;        // [256, 128]
    float* gcnt     = gsum + N_GRAPHS * NF; // [256]
    float* Wp0      = gcnt + N_GRAPHS;      // [128,128] packed
    float* Wp1      = Wp0 + NF * NF;        // [128,128] packed

    const int TB = 256;
    const int fillBlocksNE   = (NE + TB - 1) / TB;
    const int scatterBlocks  = (N_EDGES * 32) / TB;
    const int gemmBlocks     = N_NODES / MT;          // 3125, exact
    const int statsBlocks    = 400;
    const int poolBlocks     = (N_NODES * 32 + TB - 1) / TB;
    const int packBlocks     = ((NF / 2) * NF + TB - 1) / TB;  // 32

    const float* cur = x;
    for (int layer = 0; layer < 3; ++layer) {
        int relu = (layer < 2) ? 1 : 0;
        fill_kernel<<<fillBlocksNE, TB, 0, stream>>>(agg, NE, 0.0f);
        scatter_kernel<<<scatterBlocks, TB, 0, stream>>>(cur, src, dst, agg);
        pack_kernel<<<packBlocks, TB, 0, stream>>>(Wr[layer], Wp0);
        pack_kernel<<<packBlocks, TB, 0, stream>>>(Wroot[layer], Wp1);
        fill_kernel<<<1, TB, 0, stream>>>(colsum, NF, 0.0f);
        fill_kernel<<<1, TB, 0, stream>>>(colsumsq, NF, 0.0f);
        gemm_kernel<<<gemmBlocks, TB, 0, stream>>>(agg, cur, Wp0, Wp1,
                                                   bb[layer], h, relu);
        stats_kernel<<<statsBlocks, TB, 0, stream>>>(h, colsum, colsumsq);
        bn_kernel<<<fillBlocksNE, TB, 0, stream>>>(h, colsum, colsumsq,
                                                   gm[layer], be[layer], hn);
        cur = hn;
    }

    fill_kernel<<<(N_GRAPHS * NF + TB - 1) / TB, TB, 0, stream>>>(gsum, N_GRAPHS * NF, 0.0f);
    fill_kernel<<<1, TB, 0, stream>>>(gcnt, N_GRAPHS, 0.0f);
    pool_kernel<<<poolBlocks, TB, 0, stream>>>(hn, batch, gsum, gcnt);
    final_kernel<<<1, TB, 0, stream>>>(gsum, gcnt, Wlin, blin, out);
}